// Potential_6347961663538
// MI455X (gfx1250) — compile-verified
//
#include <hip/hip_runtime.h>

// CDNA5 / gfx1250. wave32. WMMA F32 16x16x4 computes pairwise squared
// distances for 16 pixels x 16 atoms per instruction via the rank-4
// factorization  d2 = [gx,gy,|g|^2,1] . [-2cx,-2cy,1,|c|^2]^T.
// exp() terms go through raw v_exp_f32 (exp2 with pre-folded log2e scale);
// accumulation forced to v_fma via __builtin_fmaf.

typedef float v2f __attribute__((ext_vector_type(2)));
typedef float v8f __attribute__((ext_vector_type(8)));

#define SIDELEN 128
#define NATOMS  2048
#define NTILES  (NATOMS / 16)
#define TILE_F  208                      // floats per 16-atom LDS block
#define SMEM_BYTES (NTILES * TILE_F * 4) // 106496 B of the 320KB WGP LDS

__global__ __launch_bounds__(256) void potential_kernel(
    const float* __restrict__ coords,   // [4,2048,3]
    const float* __restrict__ ffa,      // [2048,5]
    const float* __restrict__ ffb,      // [2048,5]
    float* __restrict__ out)            // [4,128,128]
{
    extern __shared__ float smem[];
    const int   tid   = threadIdx.x;
    const int   b     = blockIdx.y;
    const float kPI   = 3.14159265358979323846f;
    const float LOG2E = 1.4426950408889634f;

    // ---- stage derived per-atom data into LDS (each WG stages all atoms) ----
    // Per 16-atom tile block (208 floats):
    //   [0..15]    -2*cx            [16..31]  -2*cy        [32..47] cx^2+cy^2
    //   [48..127]  s[f][n]  = -pi*invb*log2e   (exp2 scale)
    //   [128..207] coef[f][n] = ff_a*invb
    for (int a = tid; a < NATOMS; a += 256) {
        float cx = coords[(b * NATOMS + a) * 3 + 0];
        float cy = coords[(b * NATOMS + a) * 3 + 1];
        int t = a >> 4, n = a & 15;
        float* blk = &smem[t * TILE_F];
        blk[n]      = -2.0f * cx;
        blk[16 + n] = -2.0f * cy;
        blk[32 + n] = cx * cx + cy * cy;
#pragma unroll
        for (int f = 0; f < 5; ++f) {
            float invb = (4.0f * kPI) / ffb[a * 5 + f];
            blk[48  + f * 16 + n] = -kPI * invb * LOG2E;
            blk[128 + f * 16 + n] = ffa[a * 5 + f] * invb;
        }
    }
    __syncthreads();

    // ---- per-wave 16-pixel tile ----
    const int lane = tid & 31;
    const int wave = tid >> 5;
    const int h    = lane >> 4;   // which 16-lane half
    const int n    = lane & 15;   // A-row M / B-column N index for this lane
    const int tile = blockIdx.x * 8 + wave;       // 0..1023 (16-pixel tiles)

    // grid value for pixel p = tile*16 + n :
    //   component0 = ax[p & 127], component1 = ax[p >> 7], ax[i] = i - 64
    const int   p  = tile * 16 + n;
    const float gx = (float)((p & 127) - 64);
    const float gy = (float)((p >> 7) - 64);

    // A-matrix 16x4 f32 layout: lanes0-15 V0=K0,V1=K1 ; lanes16-31 V0=K2,V1=K3
    v2f amat;
    amat.x = h ? (gx * gx + gy * gy) : gx;   // K0 = gx,   K2 = |g|^2
    amat.y = h ? 1.0f                : gy;   // K1 = gy,   K3 = 1

    float acc[8];
#pragma unroll
    for (int r = 0; r < 8; ++r) acc[r] = 0.0f;

    const v8f    czero = {};
    const float* blk   = &smem[n];
    for (int t = 0; t < NTILES; ++t, blk += TILE_F) {
        float p0 = blk[0];            // -2*cx[n]
        float p1 = blk[16];           // -2*cy[n]
        float pw = blk[32];           // |c[n]|^2
        // B-matrix 4x16 f32 (mirror of A striping):
        //   V0: lanes0-15 K0, lanes16-31 K2 ; V1: lanes0-15 K1, lanes16-31 K3
        v2f bmat;
        bmat.x = h ? 1.0f : p0;       // K0 = -2cx, K2 = 1
        bmat.y = h ? pw   : p1;       // K1 = -2cy, K3 = |c|^2

        // D[m][n] = squared distance pixel m <-> atom n (matrix pipe)
        v8f d2 = __builtin_amdgcn_wmma_f32_16x16x4_f32(
            false, amat, false, bmat, (short)0, czero, false, false);

        float s0 = blk[48],  s1 = blk[64],  s2 = blk[80],
              s3 = blk[96],  s4 = blk[112];
        float c0 = blk[128], c1 = blk[144], c2 = blk[160],
              c3 = blk[176], c4 = blk[192];
#pragma unroll
        for (int r = 0; r < 8; ++r) {
            float v = d2[r];
            acc[r] = __builtin_fmaf(c0, __builtin_amdgcn_exp2f(s0 * v), acc[r]);
            acc[r] = __builtin_fmaf(c1, __builtin_amdgcn_exp2f(s1 * v), acc[r]);
            acc[r] = __builtin_fmaf(c2, __builtin_amdgcn_exp2f(s2 * v), acc[r]);
            acc[r] = __builtin_fmaf(c3, __builtin_amdgcn_exp2f(s3 * v), acc[r]);
            acc[r] = __builtin_fmaf(c4, __builtin_amdgcn_exp2f(s4 * v), acc[r]);
        }
    }

    // Sum over the 16 atom columns of each half (masks <16 stay in-half, wave32).
#pragma unroll
    for (int r = 0; r < 8; ++r) {
        float v = acc[r];
        v += __shfl_xor(v, 1, 32);
        v += __shfl_xor(v, 2, 32);
        v += __shfl_xor(v, 4, 32);
        v += __shfl_xor(v, 8, 32);
        acc[r] = v;
    }

    // D layout: lanes0-15 VGPR r -> pixel M=r ; lanes16-31 -> M=r+8.
    float* ob = &out[b * (SIDELEN * SIDELEN) + tile * 16 + h * 8];
#pragma unroll
    for (int r = 0; r < 8; ++r)
        if (n == r) ob[r] = acc[r];
}

extern "C" void kernel_launch(void* const* d_in, const int* in_sizes, int n_in,
                              void* d_out, int out_size, void* d_ws, size_t ws_size,
                              hipStream_t stream) {
    (void)in_sizes; (void)n_in; (void)d_ws; (void)ws_size; (void)out_size;
    const float* coords = (const float*)d_in[0];   // [4,2048,3]
    const float* ffa    = (const float*)d_in[1];   // [2048,5]
    const float* ffb    = (const float*)d_in[2];   // [2048,5]
    float*       out    = (float*)d_out;           // [4,1,128,128]

    static bool attr_set = false;
    if (!attr_set) {
        hipFuncSetAttribute((const void*)potential_kernel,
                            hipFuncAttributeMaxDynamicSharedMemorySize, SMEM_BYTES);
        attr_set = true;
    }

    dim3 grid(128, 4, 1);   // 128 pixel-tile groups x 4 batches
    dim3 block(256, 1, 1);  // 8 waves; each wave owns one 16-pixel tile
    potential_kernel<<<grid, block, SMEM_BYTES, stream>>>(coords, ffa, ffb, out);
}